// Encoder_36953898615065
// MI455X (gfx1250) — compile-verified
//
#include <hip/hip_runtime.h>

// ---------------------------------------------------------------------------
// GENConv x3 for MI455X (gfx1250, wave32).
// Memory-bound: 3 edge passes x ~600 MB of gather+atomic traffic >> 10 GFLOP
// of GEMM, so keep fp32 and use V_WMMA_F32_16X16X4_F32 for exact tensor-core
// GEMMs. Each wave computes a 16x64 output strip: one A fragment feeds 4
// WMMAs (4x A-traffic reduction, 4x wmma density per load).
// ---------------------------------------------------------------------------

typedef __attribute__((ext_vector_type(2))) float v2f;
typedef __attribute__((ext_vector_type(8))) float v8f;

#define DFE 128     // feature dim D
#define DH  256     // hidden dim 2D
#define EPS_MSG 1e-7f
#define EPS_BN  1e-5f

// ---------------- utility kernels ----------------
__global__ __launch_bounds__(256) void zero_f32(float* __restrict__ p, long long n) {
  long long i = (long long)blockIdx.x * blockDim.x + threadIdx.x;
  if (i < n) p[i] = 0.0f;
}

__global__ __launch_bounds__(256) void copy_f32(float* __restrict__ d,
                                                const float* __restrict__ s,
                                                long long n) {
  long long i = (long long)blockIdx.x * blockDim.x + threadIdx.x;
  if (i < n) d[i] = s[i];
}

// ---------------- edge passes: one wave per edge, float4 per lane ----------
__device__ __forceinline__ float4 load_msg(const float* __restrict__ x,
                                           long long s, int d) {
  const float4 xv = *(const float4*)(x + s * DFE + d);
  float4 m;
  m.x = fmaxf(xv.x, 0.0f) + EPS_MSG;
  m.y = fmaxf(xv.y, 0.0f) + EPS_MSG;
  m.z = fmaxf(xv.z, 0.0f) + EPS_MSG;
  m.w = fmaxf(xv.w, 0.0f) + EPS_MSG;
  return m;
}

__global__ __launch_bounds__(256) void edge_max(const float* __restrict__ x,
                                                const int* __restrict__ src,
                                                const int* __restrict__ dst,
                                                float* __restrict__ mmax, int E) {
  long long idx = (long long)blockIdx.x * 256 + threadIdx.x;
  int e = (int)(idx >> 5);
  if (e >= E) return;
  int d = ((int)idx & 31) << 2;
  float4 m = load_msg(x, (long long)src[e], d);
  int* mp = (int*)(mmax + (long long)dst[e] * DFE + d);
  // msg > 0 always, so int-punned compare is monotonic; mmax init to 0.0f.
  atomicMax(mp + 0, __float_as_int(m.x));
  atomicMax(mp + 1, __float_as_int(m.y));
  atomicMax(mp + 2, __float_as_int(m.z));
  atomicMax(mp + 3, __float_as_int(m.w));
}

__global__ __launch_bounds__(256) void edge_denom(const float* __restrict__ x,
                                                  const int* __restrict__ src,
                                                  const int* __restrict__ dst,
                                                  const float* __restrict__ mmax,
                                                  float* __restrict__ denom, int E) {
  long long idx = (long long)blockIdx.x * 256 + threadIdx.x;
  int e = (int)(idx >> 5);
  if (e >= E) return;
  int d = ((int)idx & 31) << 2;
  long long t = (long long)dst[e];
  float4 m = load_msg(x, (long long)src[e], d);
  const float4 mv = *(const float4*)(mmax + t * DFE + d);
  float* dp = denom + t * DFE + d;
  atomicAdd(dp + 0, __expf(m.x - mv.x));
  atomicAdd(dp + 1, __expf(m.y - mv.y));
  atomicAdd(dp + 2, __expf(m.z - mv.z));
  atomicAdd(dp + 3, __expf(m.w - mv.w));
}

__global__ __launch_bounds__(256) void edge_agg(const float* __restrict__ x,
                                                const int* __restrict__ src,
                                                const int* __restrict__ dst,
                                                const float* __restrict__ mmax,
                                                const float* __restrict__ denom,
                                                float* __restrict__ agg, int E) {
  long long idx = (long long)blockIdx.x * 256 + threadIdx.x;
  int e = (int)(idx >> 5);
  if (e >= E) return;
  int d = ((int)idx & 31) << 2;
  long long t = (long long)dst[e];
  float4 m = load_msg(x, (long long)src[e], d);
  const float4 mv = *(const float4*)(mmax + t * DFE + d);
  const float4 dv = *(const float4*)(denom + t * DFE + d);
  float* op = agg + t * DFE + d;
  atomicAdd(op + 0, m.x * (__expf(m.x - mv.x) / dv.x));
  atomicAdd(op + 1, m.y * (__expf(m.y - mv.y) / dv.y));
  atomicAdd(op + 2, m.z * (__expf(m.z - mv.z) / dv.z));
  atomicAdd(op + 3, m.w * (__expf(m.w - mv.w) / dv.w));
}

// ---------------- WMMA fp32 GEMM: C[M,NCOL] = A[M,KDIM] @ B[KDIM,NCOL] + bias
// One 16x64 output strip per wave (4 adjacent 16x16 tiles): the A fragment is
// loaded once per k-step and feeds 4 V_WMMA_F32_16X16X4_F32 ops.
// A frag (ISA 7.12.2): lanes 0-15 -> M=lane, K={k,k+1}; lanes 16-31 -> K={k+2,k+3}
// B frag: lanes 0-15 -> row k (cols n..n+15) / row k+1; lanes 16-31 -> k+2/k+3
// C frag: VGPR v -> row v (lanes<16) or v+8 (lanes>=16); col = lane&15
template <int NCOL, int KDIM, bool BN_IN, bool RELU_OUT>
__global__ __launch_bounds__(256) void gemm_wmma(const float* __restrict__ A,
                                                 const float* __restrict__ B,
                                                 const float* __restrict__ bias,
                                                 const float* __restrict__ scale,
                                                 const float* __restrict__ shift,
                                                 float* __restrict__ C, int M) {
  const int lane = threadIdx.x & 31;
  const int wave = threadIdx.x >> 5;
  const int strips_n = NCOL / 64;            // 4 tiles per strip
  const int tiles_m = (M + 15) >> 4;
  const int strip = blockIdx.x * 8 + wave;
  if (strip >= tiles_m * strips_n) return;
  const int tm = strip / strips_n;
  const int sn = strip - tm * strips_n;
  const int col   = lane & 15;          // N index within tile / A row within tile
  const int khalf = (lane >> 4) << 1;   // 0 for lanes 0-15, 2 for lanes 16-31
  const int row0 = tm << 4;
  const int n0   = sn << 6;             // strip covers cols n0..n0+63
  int arow = row0 + col;
  if (arow >= M) arow = M - 1;          // clamp (stores are guarded below)

  const float* ap = A + (long long)arow * KDIM + khalf;
  const float* bp = B + (long long)khalf * NCOL + n0 + col;

  v8f acc0 = {}, acc1 = {}, acc2 = {}, acc3 = {};
  for (int k = 0; k < KDIM; k += 4) {
    float a0 = ap[k];
    float a1 = ap[k + 1];
    if (BN_IN) {
      a0 = fmaxf(a0 * scale[k + khalf]     + shift[k + khalf],     0.0f);
      a1 = fmaxf(a1 * scale[k + khalf + 1] + shift[k + khalf + 1], 0.0f);
    }
    v2f a;
    a.x = a0;
    a.y = a1;
    const float* bk0 = bp + (long long)k * NCOL;
    const float* bk1 = bk0 + NCOL;
    v2f b0, b1, b2, b3;
    b0.x = bk0[0];  b0.y = bk1[0];
    b1.x = bk0[16]; b1.y = bk1[16];
    b2.x = bk0[32]; b2.y = bk1[32];
    b3.x = bk0[48]; b3.y = bk1[48];
    acc0 = __builtin_amdgcn_wmma_f32_16x16x4_f32(false, a, false, b0,
                                                 (short)0, acc0, false, false);
    acc1 = __builtin_amdgcn_wmma_f32_16x16x4_f32(false, a, false, b1,
                                                 (short)0, acc1, false, false);
    acc2 = __builtin_amdgcn_wmma_f32_16x16x4_f32(false, a, false, b2,
                                                 (short)0, acc2, false, false);
    acc3 = __builtin_amdgcn_wmma_f32_16x16x4_f32(false, a, false, b3,
                                                 (short)0, acc3, false, false);
  }

  const int rbase = row0 + ((lane >> 4) << 3);
  float* cp = C + (long long)rbase * NCOL + n0 + col;
  v8f accs[4] = {acc0, acc1, acc2, acc3};
#pragma unroll
  for (int j = 0; j < 4; ++j) {
    const float bv = bias[n0 + j * 16 + col];
#pragma unroll
    for (int i = 0; i < 8; ++i) {
      if (rbase + i < M) {
        float v = accs[j][i] + bv;
        if (RELU_OUT) v = fmaxf(v, 0.0f);
        cp[(long long)i * NCOL + j * 16] = v;
      }
    }
  }
}

// ---------------- BatchNorm statistics over the node axis -------------------
__global__ __launch_bounds__(256) void bn_stats(const float* __restrict__ H,
                                                float* __restrict__ sums, int M) {
  const int c = threadIdx.x;                 // one thread per column (256 cols)
  const int r0 = blockIdx.x * 128;
  int r1 = r0 + 128;
  if (r1 > M) r1 = M;
  float s = 0.0f, s2 = 0.0f;
  for (int r = r0; r < r1; ++r) {
    float v = H[(long long)r * DH + c];
    s += v;
    s2 += v * v;
  }
  atomicAdd(&sums[c], s);
  atomicAdd(&sums[DH + c], s2);
}

__global__ __launch_bounds__(256) void bn_params(const float* __restrict__ sums,
                                                 const float* __restrict__ gamma,
                                                 const float* __restrict__ beta,
                                                 float* __restrict__ scale,
                                                 float* __restrict__ shift, int M) {
  const int c = threadIdx.x;
  if (c >= DH) return;
  const float inv_m = 1.0f / (float)M;
  const float mu  = sums[c] * inv_m;
  const float var = sums[DH + c] * inv_m - mu * mu;
  const float sc  = gamma[c] * rsqrtf(var + EPS_BN);
  scale[c] = sc;
  shift[c] = beta[c] - mu * sc;
}

// ---------------------------------------------------------------------------
extern "C" void kernel_launch(void* const* d_in, const int* in_sizes, int n_in,
                              void* d_out, int out_size, void* d_ws, size_t ws_size,
                              hipStream_t stream) {
  // setup_inputs order: x, W1, b1, gamma, beta, W2, b2, src, dst
  const float* x     = (const float*)d_in[0];
  const float* W1    = (const float*)d_in[1];
  const float* b1    = (const float*)d_in[2];
  const float* gamma = (const float*)d_in[3];
  const float* beta  = (const float*)d_in[4];
  const float* W2    = (const float*)d_in[5];
  const float* b2    = (const float*)d_in[6];
  const int*   src   = (const int*)d_in[7];
  const int*   dst   = (const int*)d_in[8];
  float* out = (float*)d_out;

  const int N = in_sizes[0] / DFE;
  const int E = in_sizes[7];
  const long long nd = (long long)N * DFE;

  // workspace layout (floats)
  float* ws    = (float*)d_ws;
  float* mmax  = ws;                 // [N,D]
  float* denom = mmax + nd;          // [N,D]
  float* agg   = denom + nd;         // [N,D]
  float* h     = agg + nd;           // [N,2D]
  float* sums  = h + 2 * nd;         // [2*2D]
  float* scale = sums + 2 * DH;      // [2D]
  float* shift = scale + DH;         // [2D]
  float* xb0   = shift + DH;         // [N,D]
  float* xb1   = xb0 + nd;           // [N,D]

  const dim3 b256(256);
  const int nd_blocks   = (int)((nd + 255) / 256);
  const long long et    = (long long)E * 32;
  const int edge_blocks = (int)((et + 255) / 256);
  const int strips1 = ((N + 15) / 16) * (DH / 64);
  const int strips2 = ((N + 15) / 16) * (DFE / 64);
  const int g1 = (strips1 + 7) / 8;
  const int g2 = (strips2 + 7) / 8;
  const int bn_blocks = (N + 127) / 128;

  const float* xin = x;
  for (int i = 0; i < 3; ++i) {
    float* xout = (i == 0) ? xb0 : (i == 1) ? xb1 : out;
    const float* W1i = W1 + (long long)i * DFE * DH;
    const float* b1i = b1 + i * DH;
    const float* gi  = gamma + i * DH;
    const float* bei = beta + i * DH;
    const float* W2i = W2 + (long long)i * DH * DFE;
    const float* b2i = b2 + i * DFE;

    zero_f32<<<nd_blocks, b256, 0, stream>>>(mmax, nd);
    zero_f32<<<nd_blocks, b256, 0, stream>>>(denom, nd);
    copy_f32<<<nd_blocks, b256, 0, stream>>>(agg, xin, nd);   // residual seed
    zero_f32<<<1, b256, 0, stream>>>(sums, 2 * DH);

    edge_max  <<<edge_blocks, b256, 0, stream>>>(xin, src, dst, mmax, E);
    edge_denom<<<edge_blocks, b256, 0, stream>>>(xin, src, dst, mmax, denom, E);
    edge_agg  <<<edge_blocks, b256, 0, stream>>>(xin, src, dst, mmax, denom, agg, E);

    gemm_wmma<DH, DFE, false, false><<<g1, b256, 0, stream>>>(
        agg, W1i, b1i, nullptr, nullptr, h, N);
    bn_stats <<<bn_blocks, b256, 0, stream>>>(h, sums, N);
    bn_params<<<1, b256, 0, stream>>>(sums, gi, bei, scale, shift, N);
    gemm_wmma<DFE, DH, true, true><<<g2, b256, 0, stream>>>(
        h, W2i, b2i, scale, shift, xout, N);

    xin = xout;
  }
}